// ActionHeadGMM_57071525429810
// MI455X (gfx1250) — compile-verified
//
#include <hip/hip_runtime.h>

typedef __attribute__((ext_vector_type(4))) float v4f;
typedef __attribute__((ext_vector_type(2))) float v2f;
typedef __attribute__((ext_vector_type(8))) float v8f;

#define LOG2PI 1.8378770664093453f
#define LOG5   1.6094379124341003f

// Kernel 1: streaming NLL partial sums. One batch element per grid-stride
// iteration; 127 floats in, 1 float contribution out. HBM-bound: all bulk
// loads are 128-bit non-temporal (data is touched exactly once, > L2 size).
__global__ __launch_bounds__(256) void gmm_partial_kernel(
    const float* __restrict__ means, const float* __restrict__ covs,
    const float* __restrict__ mix,   const float* __restrict__ tgt,
    float* __restrict__ partials, int Bn) {
  __shared__ float sdata[256];
  const int tid = threadIdx.x;
  const long long stride = (long long)gridDim.x * blockDim.x;
  float acc = 0.0f;

  for (long long b = (long long)blockIdx.x * blockDim.x + tid; b < Bn; b += stride) {
    // action targets: 7 floats, 28B stride (not 16B aligned) -> scalar loads
    const float* tp = tgt + b * 7;
    float t[7];
#pragma unroll
    for (int a = 0; a < 7; ++a) t[a] = tp[a];

    // mixing logits: 8 floats, 32B stride -> two b128 loads
    const v4f* xv = (const v4f*)(mix + b * 8);
    v4f x0 = __builtin_nontemporal_load(xv);
    v4f x1 = __builtin_nontemporal_load(xv + 1);
    float xs[8] = {x0.x, x0.y, x0.z, x0.w, x1.x, x1.y, x1.z, x1.w};
    float xmax = xs[0];
#pragma unroll
    for (int k = 1; k < 8; ++k) xmax = fmaxf(xmax, xs[k]);
    float e[8];
    float esum = 0.0f;
#pragma unroll
    for (int k = 0; k < 8; ++k) { e[k] = __expf(xs[k] - xmax); esum += e[k]; }

    // means / covariances: 56 floats each, 224B stride -> 14 b128 loads each
    const v4f* mv = (const v4f*)(means + b * 56);
    const v4f* cv = (const v4f*)(covs  + b * 56);
    float m[56], c[56];
#pragma unroll
    for (int i = 0; i < 14; ++i) {
      v4f q = __builtin_nontemporal_load(mv + i);
      m[4*i+0] = q.x; m[4*i+1] = q.y; m[4*i+2] = q.z; m[4*i+3] = q.w;
    }
#pragma unroll
    for (int i = 0; i < 14; ++i) {
      v4f q = __builtin_nontemporal_load(cv + i);
      c[4*i+0] = q.x; c[4*i+1] = q.y; c[4*i+2] = q.z; c[4*i+3] = q.w;
    }

    // S_k = sum_a [ log5 - log(1+e^{-c}) + d^2 * (1+e^{-c}) / 5 ]
    // loss_b = 0.5*(7*log2pi) + 0.5 * (sum_k e_k*S_k) / esum
    float S = 0.0f;
#pragma unroll
    for (int k = 0; k < 8; ++k) {
      float sk = 0.0f;
#pragma unroll
      for (int a = 0; a < 7; ++a) {
        const int i = k * 7 + a;
        const float d  = t[a] - m[i];
        const float em = __expf(-c[i]);          // e^{-c}
        sk += (LOG5 - __logf(1.0f + em)) + d * d * (1.0f + em) * 0.2f;
      }
      S += e[k] * sk;
    }
    acc += 0.5f * (7.0f * LOG2PI + S / esum);
  }

  sdata[tid] = acc;
  __syncthreads();
#pragma unroll
  for (int s = 128; s > 0; s >>= 1) {
    if (tid < s) sdata[tid] += sdata[tid + s];
    __syncthreads();
  }
  if (tid == 0) partials[blockIdx.x] = sdata[0];
}

// Kernel 2: one wave32. Each lane strided-sums the block partials, then the
// 32 lane sums are reduced with V_WMMA_F32_16X16X4_F32:
//   A (16x4): VGPR0 lane m (<16) -> A[m,0]=s_m ; lane m+16 -> A[m,2]=s_{m+16};
//             VGPR1 (K=1,3) = 0.   B = all-ones (layout independent).
//   => D[m,n] = s_m + s_{m+16} for every n.
//   Lane 0 sums its 8 acc VGPRs = rows 0..7; lane 16 = rows 8..15.
__global__ __launch_bounds__(32) void gmm_finalize_kernel(
    const float* __restrict__ partials, int n, float* __restrict__ out, float scale) {
  const int lane = threadIdx.x;
  float s = 0.0f;
  for (int i = lane; i < n; i += 32) s += partials[i];

  v2f a;  a.x = s;    a.y = 0.0f;
  v2f bo; bo.x = 1.0f; bo.y = 1.0f;
  v8f cz = {};
  v8f d = __builtin_amdgcn_wmma_f32_16x16x4_f32(
      /*neg_a=*/false, a, /*neg_b=*/false, bo,
      /*c_mod=*/(short)0, cz, /*reuse_a=*/false, /*reuse_b=*/false);

  float r = 0.0f;
#pragma unroll
  for (int i = 0; i < 8; ++i) r += d[i];
  const float total = __shfl(r, 0, 32) + __shfl(r, 16, 32);
  if (lane == 0) out[0] = total * scale;
}

extern "C" void kernel_launch(void* const* d_in, const int* in_sizes, int n_in,
                              void* d_out, int out_size, void* d_ws, size_t ws_size,
                              hipStream_t stream) {
  const float* means = (const float*)d_in[0];  // [B, 8, 7]
  const float* covs  = (const float*)d_in[1];  // [B, 8, 7]
  const float* mix   = (const float*)d_in[2];  // [B, 8]
  const float* tgt   = (const float*)d_in[3];  // [B, 7]
  float* out = (float*)d_out;

  const int Bn = in_sizes[2] / 8;

  int nblocks = 2048;
  if (ws_size < (size_t)nblocks * sizeof(float)) {
    nblocks = (int)(ws_size / sizeof(float));
    if (nblocks < 1) nblocks = 1;
  }
  float* partials = (float*)d_ws;

  gmm_partial_kernel<<<nblocks, 256, 0, stream>>>(means, covs, mix, tgt, partials, Bn);
  gmm_finalize_kernel<<<1, 32, 0, stream>>>(partials, nblocks, out, 1.0f / (float)Bn);
}